// Geometric_BA_89258010346205
// MI455X (gfx1250) — compile-verified
//
#include <hip/hip_runtime.h>
#include <math.h>

// ---------------------------------------------------------------------------
// Geometric bundle adjustment (LM) for MI455X / gfx1250.
// fp32 Gram reductions (J^T W J etc.) are done with V_WMMA_F32_16X16X4_F32
// symmetric rank-4 updates (SRC0 == SRC1 == A, since B = A^T occupies the
// identical lane/VGPR placement for the f32 16x16x4 shape).
// ---------------------------------------------------------------------------

typedef float v2f __attribute__((ext_vector_type(2)));
typedef float v8f __attribute__((ext_vector_type(8)));

#define NUM_ITER 4
#define TPB 256
#define WAVES 8
#define PTS_PER_LANE 8           // points per lane per kReduce block
#define PTS_PER_BLOCK (TPB * PTS_PER_LANE)   // 2048

struct Core {
    float a0[7], a1[7];   // residual-row vectors (w*Jp | e) for Bm/g Gram
    float bb[7];          // (sqrt(cinv)*Ev | sqrt(cinv)*gd) for Hsc/bsc Gram
    float Ev[6];          // E per point (undamped)
    float cinv;           // 1/max(C*(1+L),1)
    float gd;             // g_disp per point (= Wv)
};

// Per-point LM core: projection, Jacobians, Huber weights, per-point
// marginalization quantities. Matches reference _lm_step exactly.
__device__ __forceinline__ void core_point(
    const float* __restrict__ Km, const float* __restrict__ Kim,
    const float* __restrict__ P,   // current pose, 4x4 row-major
    float x1, float y1, float x2, float y2,
    float d, float m, float Lv, Core& c)
{
    float depth = 1.0f / d;                       // reference: no clamp here
    float px = (Kim[0]*x1 + Kim[1]*y1 + Kim[2]) * depth;
    float py = (Kim[3]*x1 + Kim[4]*y1 + Kim[5]) * depth;
    float pz = (Kim[6]*x1 + Kim[7]*y1 + Kim[8]) * depth;
    float X  = P[0]*px + P[1]*py + P[2]*pz  + P[3];
    float Y  = P[4]*px + P[5]*py + P[6]*pz  + P[7];
    float Zr = P[8]*px + P[9]*py + P[10]*pz + P[11];
    float qx = Km[0]*X + Km[1]*Y + Km[2]*Zr;
    float qy = Km[3]*X + Km[4]*Y + Km[5]*Zr;
    float qz = Km[6]*X + Km[7]*Y + Km[8]*Zr;
    float zc = fmaxf(qz, 0.01f);
    float izc = 1.0f / zc;
    float e0 = x2 - qx * izc;
    float e1 = y2 - qy * izc;

    float Z  = fmaxf(Zr, 1e-12f);
    float iZ = 1.0f / Z, iZ2 = iZ * iZ;
    float fx = Km[0], fy = Km[4];

    // J_pose = -rows (reference _jacobi_pose)
    float jp0[6] = { -(fx*iZ), 0.0f,  fx*X*iZ2,  fx*X*Y*iZ2,
                     -(fx + fx*X*X*iZ2),  fx*Y*iZ };
    float jp1[6] = { 0.0f, -(fy*iZ),  fy*Y*iZ2,  fy + fy*Y*Y*iZ2,
                     -(fy*X*Y*iZ2), -(fy*X*iZ) };

    // J_disp = -(J1 @ J2)   (reference _jacobi_disp_impl)
    float DD  = -1.0f / d;
    float j2x = DD * (X - P[3]);
    float j2y = DD * (Y - P[7]);
    float j2z = DD * (Z - P[11]);
    float jd0 = -(fx*iZ*j2x - fx*X*iZ2*j2z);
    float jd1 = -(fy*iZ*j2y - fy*Y*iZ2*j2z);

    // Huber weights
    float en = sqrtf(e0*e0 + e1*e1);
    float w  = sqrtf(fminf(9.0f / fmaxf(en, 1e-12f), 1.0f)) * m;
    float w2 = w * w;

#pragma unroll
    for (int i = 0; i < 6; ++i) c.Ev[i] = w2 * (jp0[i]*jd0 + jp1[i]*jd1);
    float C  = w2 * (jd0*jd0 + jd1*jd1);
    float Cd = C * (1.0f + Lv);
    c.cinv = 1.0f / fmaxf(Cd, 1.0f);
    c.gd   = -(w * (jd0*e0 + jd1*e1));

#pragma unroll
    for (int i = 0; i < 6; ++i) { c.a0[i] = w * jp0[i]; c.a1[i] = w * jp1[i]; }
    c.a0[6] = e0; c.a1[6] = e1;

    float s = sqrtf(c.cinv);
#pragma unroll
    for (int i = 0; i < 6; ++i) c.bb[i] = s * c.Ev[i];
    c.bb[6] = s * c.gd;
}

// Reference _reproject_error -> per-point L2 norm
__device__ __forceinline__ float reproj_norm(
    const float* __restrict__ P, const float* __restrict__ Km,
    const float* __restrict__ Kim,
    float x1, float y1, float x2, float y2, float d)
{
    float depth = 1.0f / fmaxf(d, 0.01f);
    float px = (Kim[0]*x1 + Kim[1]*y1 + Kim[2]) * depth;
    float py = (Kim[3]*x1 + Kim[4]*y1 + Kim[5]) * depth;
    float pz = (Kim[6]*x1 + Kim[7]*y1 + Kim[8]) * depth;
    float X  = P[0]*px + P[1]*py + P[2]*pz  + P[3];
    float Y  = P[4]*px + P[5]*py + P[6]*pz  + P[7];
    float Zr = P[8]*px + P[9]*py + P[10]*pz + P[11];
    float qx = Km[0]*X + Km[1]*Y + Km[2]*Zr;
    float qy = Km[3]*X + Km[4]*Y + Km[5]*Zr;
    float qz = Km[6]*X + Km[7]*Y + Km[8]*Zr;
    float zc = fmaxf(qz, 0.01f);
    float e0 = x2 - qx / zc;
    float e1 = y2 - qy / zc;
    return sqrtf(e0*e0 + e1*e1);
}

// ---------------------------------------------------------------------------
// kInit: copy disp -> d_out disp region, pose -> ws poseCur, L -> ws Lcur
// ---------------------------------------------------------------------------
__global__ void kInit(const float* __restrict__ dIn, const float* __restrict__ pIn,
                      const float* __restrict__ LIn, float* __restrict__ dOut,
                      float* __restrict__ poseCur, float* __restrict__ Lcur,
                      int BN, int B)
{
    int i = blockIdx.x * blockDim.x + threadIdx.x;
    if (i < BN)     dOut[i]    = dIn[i];
    if (i < B * 16) poseCur[i] = pIn[i];
    if (i < B)      Lcur[i]    = LIn[i];
}

__global__ void kZero(float* __restrict__ red, float* __restrict__ rsum, int B)
{
    int i = blockIdx.x * blockDim.x + threadIdx.x;
    if (i < B * 128) red[i]  = 0.0f;
    if (i < B * 4)   rsum[i] = 0.0f;
}

// ---------------------------------------------------------------------------
// kReduce: per-point core + WMMA f32 16x16x4 symmetric rank-4 Gram updates.
//   acc1 top-left 7x7 accumulates [ Sum w2 Jp^T Jp  |  Sum w Jp^T e ]
//   acc2 top-left 7x7 accumulates [ Sum cinv E E^T  |  Sum cinv E gd ]
// ---------------------------------------------------------------------------
__global__ void __launch_bounds__(TPB) kReduce(
    const float* __restrict__ pts, const float* __restrict__ disp,
    const float* __restrict__ mask, const float* __restrict__ Kg,
    const float* __restrict__ Kig, const float* __restrict__ poseCur,
    const float* __restrict__ Lcur, float* __restrict__ red, int N)
{
    __shared__ float lds[WAVES * 96 * 16];   // per wave: 64 A-rows + 32 B-rows, 16f stride

    const int b    = blockIdx.y;
    const int wave = threadIdx.x >> 5;
    const int lane = threadIdx.x & 31;
    const float* Km  = Kg  + (size_t)b * 9;
    const float* Kim = Kig + (size_t)b * 9;
    const float* Pc  = poseCur + (size_t)b * 16;
    const float Lv   = Lcur[b];

    float* Arow = &lds[wave * 96 * 16];
    float* Brow = Arow + 64 * 16;

    // zero LDS once: cols 7..15 stay zero so WMMA rows 7..15 are zero
    for (int i = threadIdx.x; i < WAVES * 96 * 16; i += TPB) lds[i] = 0.0f;
    __syncthreads();

    v8f acc1 = {0.f,0.f,0.f,0.f,0.f,0.f,0.f,0.f};
    v8f acc2 = {0.f,0.f,0.f,0.f,0.f,0.f,0.f,0.f};

    const int base = (blockIdx.x * WAVES + wave) * (32 * PTS_PER_LANE);
    const int hh   = lane >> 4;      // K-half select for WMMA A layout
    const int mcol = lane & 15;      // M (feature) index for WMMA A layout

    for (int t = 0; t < PTS_PER_LANE; ++t) {
        int p = base + t * 32 + lane;
        Core c;
        if (p < N) {
            size_t o4 = (size_t)b * 4 * N;
            float x1 = pts[o4 + p];
            float y1 = pts[o4 + (size_t)N + p];
            float x2 = pts[o4 + (size_t)2 * N + p];
            float y2 = pts[o4 + (size_t)3 * N + p];
            float d  = disp[(size_t)b * N + p];
            float m  = mask[(size_t)b * N + p];
            core_point(Km, Kim, Pc, x1, y1, x2, y2, d, m, Lv, c);
        } else {
#pragma unroll
            for (int i = 0; i < 7; ++i) { c.a0[i] = 0.f; c.a1[i] = 0.f; c.bb[i] = 0.f; }
        }
#pragma unroll
        for (int i = 0; i < 7; ++i) {
            Arow[(2 * lane + 0) * 16 + i] = c.a0[i];
            Arow[(2 * lane + 1) * 16 + i] = c.a1[i];
            Brow[lane * 16 + i]           = c.bb[i];
        }
        __syncthreads();

        // Bm/g Gram: 64 residual rows -> 16 K=4 steps
#pragma unroll
        for (int k = 0; k < 16; ++k) {
            v2f av;
            av.x = Arow[(k * 4 + hh * 2 + 0) * 16 + mcol];
            av.y = Arow[(k * 4 + hh * 2 + 1) * 16 + mcol];
            acc1 = __builtin_amdgcn_wmma_f32_16x16x4_f32(
                false, av, false, av, (short)0, acc1, false, false);
        }
        // Hsc/bsc Gram: 32 point rows -> 8 K=4 steps
#pragma unroll
        for (int k = 0; k < 8; ++k) {
            v2f bv;
            bv.x = Brow[(k * 4 + hh * 2 + 0) * 16 + mcol];
            bv.y = Brow[(k * 4 + hh * 2 + 1) * 16 + mcol];
            acc2 = __builtin_amdgcn_wmma_f32_16x16x4_f32(
                false, bv, false, bv, (short)0, acc2, false, false);
        }
        __syncthreads();
    }

    // C/D layout: lane j<16, VGPR i holds C[i][j]. We need the 7x7 block.
    if (lane < 7) {
        float* r1 = red + (size_t)b * 128;
        float* r2 = r1 + 64;
#pragma unroll
        for (int i = 0; i < 7; ++i) {
            atomicAdd(&r1[i * 7 + lane], acc1[i]);
            atomicAdd(&r2[i * 7 + lane], acc2[i]);
        }
    }
}

// ---------------------------------------------------------------------------
// kSolve: per-batch damped Schur solve + SE3 exponential + pose composition.
// ---------------------------------------------------------------------------
__global__ void kSolve(const float* __restrict__ red,
                       const float* __restrict__ poseCur,
                       float* __restrict__ poseNew, float* __restrict__ dpose,
                       const float* __restrict__ Lcur, int N)
{
    int b = blockIdx.x;
    const float* r1 = red + (size_t)b * 128;
    const float* r2 = r1 + 64;
    float Lv   = Lcur[b];
    float invL = 1.0f / (1.0f + Lv);
    float Ninv = 1.0f / (float)N;

    float left[6][6], right[6];
    for (int i = 0; i < 6; ++i) {
        for (int j = 0; j < 6; ++j) {
            float Bm = r1[i * 7 + j] * Ninv;
            if (i == j) Bm = Bm * (1.0f + Lv) + 1.0f;   // diag damping + I
            float Hs = r2[i * 7 + j] * Ninv;
            left[i][j] = Bm - Hs * invL;
        }
        right[i] = -r1[i * 7 + 6] * Ninv - (r2[i * 7 + 6] * Ninv) * invL;
    }
    float sD[6];
    for (int i = 0; i < 6; ++i) sD[i] = rsqrtf(left[i][i] + 10.0f);

    float M[6][7];
    for (int i = 0; i < 6; ++i) {
        for (int j = 0; j < 6; ++j) M[i][j] = sD[i] * left[i][j] * sD[j];
        M[i][6] = sD[i] * right[i];
    }
    // Gauss-Jordan with partial pivoting
    for (int c = 0; c < 6; ++c) {
        int piv = c; float best = fabsf(M[c][c]);
        for (int r = c + 1; r < 6; ++r) {
            float v = fabsf(M[r][c]);
            if (v > best) { best = v; piv = r; }
        }
        if (piv != c)
            for (int j = 0; j < 7; ++j) { float tt = M[c][j]; M[c][j] = M[piv][j]; M[piv][j] = tt; }
        float inv = 1.0f / M[c][c];
        for (int r = 0; r < 6; ++r) {
            if (r == c) continue;
            float f = M[r][c] * inv;
            for (int j = c; j < 7; ++j) M[r][j] -= f * M[c][j];
        }
    }
    float dp[6];
    for (int i = 0; i < 6; ++i) dp[i] = sD[i] * (M[i][6] / M[i][i]);
    for (int i = 0; i < 6; ++i) dpose[(size_t)b * 8 + i] = dp[i];

    // SE3 exponential (reference _se3_pose)
    float t0 = dp[0], t1 = dp[1], t2 = dp[2];
    float w0 = dp[3], w1 = dp[4], w2 = dp[5];
    float Nr = fmaxf(sqrtf(w0*w0 + w1*w1 + w2*w2), 1e-12f);
    float r0 = w0 / Nr, rr1 = w1 / Nr, rr2 = w2 / Nr;
    float cc = cosf(Nr), sn = sinf(Nr);
    float dot = r0*r0 + rr1*rr1 + rr2*rr2;

    float R[3][3];
    R[0][0] = cc + (1.f-cc)*r0*r0;          R[0][1] = (1.f-cc)*r0*rr1 - sn*rr2;  R[0][2] = (1.f-cc)*r0*rr2 + sn*rr1;
    R[1][0] = (1.f-cc)*rr1*r0 + sn*rr2;     R[1][1] = cc + (1.f-cc)*rr1*rr1;     R[1][2] = (1.f-cc)*rr1*rr2 - sn*r0;
    R[2][0] = (1.f-cc)*rr2*r0 - sn*rr1;     R[2][1] = (1.f-cc)*rr2*rr1 + sn*r0;  R[2][2] = cc + (1.f-cc)*rr2*rr2;

    float k1 = (Nr - sn) / Nr, k2 = (1.f - cc) / Nr;
    // rx^2 = r r^T - (r.r) I  (exact for any r)
    float J[3][3];
    J[0][0] = 1.f + k1*(r0*r0 - dot);   J[0][1] = k1*r0*rr1 - k2*rr2;       J[0][2] = k1*r0*rr2 + k2*rr1;
    J[1][0] = k1*rr1*r0 + k2*rr2;       J[1][1] = 1.f + k1*(rr1*rr1 - dot); J[1][2] = k1*rr1*rr2 - k2*r0;
    J[2][0] = k1*rr2*r0 - k2*rr1;       J[2][1] = k1*rr2*rr1 + k2*r0;       J[2][2] = 1.f + k1*(rr2*rr2 - dot);

    float T[3];
    for (int i = 0; i < 3; ++i) T[i] = J[i][0]*t0 + J[i][1]*t1 + J[i][2]*t2;

    // poseNew = [[R,T],[0,0,0,1]] @ poseCur
    const float* Pc = poseCur + (size_t)b * 16;
    float* Pn = poseNew + (size_t)b * 16;
    for (int i = 0; i < 3; ++i)
        for (int j = 0; j < 4; ++j)
            Pn[i*4+j] = R[i][0]*Pc[j] + R[i][1]*Pc[4+j] + R[i][2]*Pc[8+j] + T[i]*Pc[12+j];
    Pn[12] = Pc[12]; Pn[13] = Pc[13]; Pn[14] = Pc[14]; Pn[15] = Pc[15];
}

// ---------------------------------------------------------------------------
// kEval: per-point disparity delta + reprojection errors for accept test.
// ---------------------------------------------------------------------------
__global__ void __launch_bounds__(TPB) kEval(
    const float* __restrict__ pts, const float* __restrict__ disp,
    const float* __restrict__ mask, const float* __restrict__ Kg,
    const float* __restrict__ Kig, const float* __restrict__ poseCur,
    const float* __restrict__ poseNew, const float* __restrict__ dpose,
    const float* __restrict__ Lcur, float* __restrict__ rsum, int N)
{
    int b = blockIdx.y;
    int p = blockIdx.x * blockDim.x + threadIdx.x;
    const float* Km  = Kg  + (size_t)b * 9;
    const float* Kim = Kig + (size_t)b * 9;
    const float* Pc  = poseCur + (size_t)b * 16;
    const float* Pn  = poseNew + (size_t)b * 16;
    float Lv = Lcur[b];
    float invL = 1.0f / (1.0f + Lv);

    float n1 = 0.f, n2 = 0.f, mm = 0.f;
    if (p < N) {
        size_t o4 = (size_t)b * 4 * N;
        float x1 = pts[o4 + p];
        float y1 = pts[o4 + (size_t)N + p];
        float x2 = pts[o4 + (size_t)2 * N + p];
        float y2 = pts[o4 + (size_t)3 * N + p];
        float d  = disp[(size_t)b * N + p];
        float m  = mask[(size_t)b * N + p];
        Core c;
        core_point(Km, Kim, Pc, x1, y1, x2, y2, d, m, Lv, c);
        const float* dp = dpose + (size_t)b * 8;
        float dotE = c.Ev[0]*dp[0] + c.Ev[1]*dp[1] + c.Ev[2]*dp[2]
                   + c.Ev[3]*dp[3] + c.Ev[4]*dp[4] + c.Ev[5]*dp[5];
        float dd = c.cinv * (c.gd - dotE * invL);
        float dn = fminf(fmaxf(d + dd, 0.01f), 10.0f);
        n1 = m * reproj_norm(Pn, Km, Kim, x1, y1, x2, y2, dn);
        n2 = m * reproj_norm(Pc, Km, Kim, x1, y1, x2, y2, d);
        mm = m;
    }
    for (int off = 16; off > 0; off >>= 1) {
        n1 += __shfl_down(n1, off, 32);
        n2 += __shfl_down(n2, off, 32);
        mm += __shfl_down(mm, off, 32);
    }
    if ((threadIdx.x & 31) == 0) {
        atomicAdd(&rsum[b * 4 + 0], n1);
        atomicAdd(&rsum[b * 4 + 1], n2);
        atomicAdd(&rsum[b * 4 + 2], mm);
    }
}

__global__ void kDecide(const float* __restrict__ rsum, const float* __restrict__ Lcur,
                        float* __restrict__ Lnext, float* __restrict__ Mflag)
{
    int b = blockIdx.x;
    float ms = rsum[b * 4 + 2];
    float r1 = rsum[b * 4 + 0] / ms;
    float r2 = rsum[b * 4 + 1] / ms;
    float M  = (r1 < r2) ? 1.0f : 0.0f;
    Mflag[b] = M;
    float L  = Lcur[b];
    float Ln = (M == 1.0f) ? L * 0.5f : L * 5.0f;
    Lnext[b] = fminf(fmaxf(Ln, 0.01f), 1000000.0f);
}

// kApply: where accepted, recompute the disparity delta (old pose/L) and
// commit the clipped new disparity into d_out.
__global__ void __launch_bounds__(TPB) kApply(
    const float* __restrict__ pts, float* __restrict__ disp,
    const float* __restrict__ mask, const float* __restrict__ Kg,
    const float* __restrict__ Kig, const float* __restrict__ poseCur,
    const float* __restrict__ dpose, const float* __restrict__ Lcur,
    const float* __restrict__ Mflag, int N)
{
    int b = blockIdx.y;
    if (Mflag[b] == 0.0f) return;
    int p = blockIdx.x * blockDim.x + threadIdx.x;
    if (p >= N) return;
    const float* Km  = Kg  + (size_t)b * 9;
    const float* Kim = Kig + (size_t)b * 9;
    const float* Pc  = poseCur + (size_t)b * 16;
    float Lv = Lcur[b];
    float invL = 1.0f / (1.0f + Lv);
    size_t o4 = (size_t)b * 4 * N;
    float x1 = pts[o4 + p];
    float y1 = pts[o4 + (size_t)N + p];
    float x2 = pts[o4 + (size_t)2 * N + p];
    float y2 = pts[o4 + (size_t)3 * N + p];
    float d  = disp[(size_t)b * N + p];
    float m  = mask[(size_t)b * N + p];
    Core c;
    core_point(Km, Kim, Pc, x1, y1, x2, y2, d, m, Lv, c);
    const float* dp = dpose + (size_t)b * 8;
    float dotE = c.Ev[0]*dp[0] + c.Ev[1]*dp[1] + c.Ev[2]*dp[2]
               + c.Ev[3]*dp[3] + c.Ev[4]*dp[4] + c.Ev[5]*dp[5];
    float dd = c.cinv * (c.gd - dotE * invL);
    disp[(size_t)b * N + p] = fminf(fmaxf(d + dd, 0.01f), 10.0f);
}

__global__ void kCommit(float* __restrict__ poseCur, const float* __restrict__ poseNew,
                        float* __restrict__ Lcur, const float* __restrict__ Lnext,
                        const float* __restrict__ Mflag)
{
    int b = blockIdx.x;
    int t = threadIdx.x;
    if (Mflag[b] != 0.0f) poseCur[(size_t)b * 16 + t] = poseNew[(size_t)b * 16 + t];
    if (t == 0) Lcur[b] = Lnext[b];
}

__global__ void kFinish(const float* __restrict__ poseCur, float* __restrict__ outPose, int n)
{
    int i = threadIdx.x;
    if (i < n) outPose[i] = poseCur[i];
}

// ---------------------------------------------------------------------------
extern "C" void kernel_launch(void* const* d_in, const int* in_sizes, int n_in,
                              void* d_out, int out_size, void* d_ws, size_t ws_size,
                              hipStream_t stream)
{
    const float* pts  = (const float*)d_in[0];   // (B,4,N)
    const float* dsp  = (const float*)d_in[1];   // (B,1,N)
    const float* pose = (const float*)d_in[2];   // (B,4,4)
    const float* Kg   = (const float*)d_in[3];   // (B,3,3)
    const float* Kig  = (const float*)d_in[4];   // (B,3,3)
    const float* msk  = (const float*)d_in[5];   // (B,1,N)
    const float* Lin  = (const float*)d_in[6];   // (B,1,1)

    const int BN = in_sizes[1];
    const int B  = in_sizes[2] / 16;
    const int N  = BN / B;

    float* out     = (float*)d_out;
    float* outPose = out;            // B*16 floats
    float* outDisp = out + B * 16;   // B*N floats

    // small workspace layout (floats)
    float* ws      = (float*)d_ws;
    float* red     = ws;                  // B*128  (C1 at +0..48, C2 at +64..112)
    float* poseCur = red + B * 128;       // B*16
    float* poseNew = poseCur + B * 16;    // B*16
    float* dpose   = poseNew + B * 16;    // B*8
    float* Lcur    = dpose + B * 8;       // B
    float* Lnext   = Lcur + B;            // B
    float* rsum    = Lnext + B;           // B*4
    float* Mflag   = rsum + B * 4;        // B

    kInit<<<(BN + TPB - 1) / TPB, TPB, 0, stream>>>(dsp, pose, Lin, outDisp,
                                                    poseCur, Lcur, BN, B);

    dim3 gr((N + PTS_PER_BLOCK - 1) / PTS_PER_BLOCK, B);
    dim3 ge((N + TPB - 1) / TPB, B);

    for (int it = 0; it < NUM_ITER; ++it) {
        kZero<<<(B * 128 + TPB - 1) / TPB, TPB, 0, stream>>>(red, rsum, B);
        kReduce<<<gr, TPB, 0, stream>>>(pts, outDisp, msk, Kg, Kig, poseCur,
                                        Lcur, red, N);
        kSolve<<<B, 1, 0, stream>>>(red, poseCur, poseNew, dpose, Lcur, N);
        kEval<<<ge, TPB, 0, stream>>>(pts, outDisp, msk, Kg, Kig, poseCur,
                                      poseNew, dpose, Lcur, rsum, N);
        kDecide<<<B, 1, 0, stream>>>(rsum, Lcur, Lnext, Mflag);
        kApply<<<ge, TPB, 0, stream>>>(pts, outDisp, msk, Kg, Kig, poseCur,
                                       dpose, Lcur, Mflag, N);
        kCommit<<<B, 16, 0, stream>>>(poseCur, poseNew, Lcur, Lnext, Mflag);
    }
    kFinish<<<1, 128, 0, stream>>>(poseCur, outPose, B * 16);
}